// ParametricDrumSynth_27565100106268
// MI455X (gfx1250) — compile-verified
//
#include <hip/hip_runtime.h>
#include <hip/hip_bf16.h>
#include <math.h>

// ParametricDrumSynth for MI455X (gfx1250, wave32).
// Single workgroup, everything LDS-resident (CDNA5: 320KB/WGP).
// One-pole IIR parallelized as a blocked scan whose dense intra-block part
// runs on V_WMMA_F32_16X16X4_F32; resonator parallelized as a wavefront
// over chunks of (floor(delay)-1) samples.

typedef __attribute__((ext_vector_type(2))) float v2f;
typedef __attribute__((ext_vector_type(8))) float v8f;

#define NS      24000          // num samples
#define SRATE   48000.0f
#define NCH     (NS / 16)      // 1500 filter chunks
#define NTILES  ((NCH + 15) / 16)  // 94 WMMA column-tiles
#define BLOCK   512
#define NWAVES  (BLOCK / 32)
#define CUTF    0.1f           // NOISE_CUTOFF (compile-time constant)
#define ADEC    0.9f           // 1 - cutoff
#define A16     0.1853020188851841f  // 0.9^16

// M[j][k] = cutoff * a^(j-k) for k<=j, else 0  (lower-triangular Toeplitz)
__device__ __forceinline__ float mcoef(int j, int k) {
    return (k <= j) ? CUTF * powf(ADEC, (float)(j - k)) : 0.0f;
}

__global__ __launch_bounds__(BLOCK)
void drum_synth_kernel(const float* __restrict__ params,
                       const float* __restrict__ noise_t,
                       const float* __restrict__ noise_n,
                       float* __restrict__ out) {
    extern __shared__ float smem[];
    float* xbuf  = smem;                 // [NS]  noise_t, later transient
    float* pbuf  = smem + NS;            // [NS]  IIR partials, later resonator out
    float* sarr  = smem + 2 * NS;        // [NCH] chunk carries
    float* scanA = sarr + NCH + 4;       // [BLOCK]
    float* scanB = scanA + BLOCK;        // [BLOCK]

    const int tid = threadIdx.x;

    const float decay_t  = params[1];
    const float freq_t   = params[2];
    const float sat      = params[3];
    const float gain_t   = params[4];
    const float freq_r   = params[5];
    const float feedback = params[6];
    const float gain_r   = params[7];
    const float attack_n = params[8];
    const float decay_n  = params[9];
    const float gain_n   = params[10];
    const float tstep    = 1.0f / (float)(NS - 1);   // linspace(0,1,NS)

    // ---- Phase 1: stage noise_t into LDS ----
    for (int i = tid; i < NS; i += BLOCK) xbuf[i] = noise_t[i];
    __syncthreads();

    // ---- Phase 2: intra-chunk IIR partials  Y = M * X  via f32 WMMA ----
    // X[k][col] = xbuf[col*16 + k]; each tile handles 16 chunks (columns).
    {
        const int w    = tid >> 5;
        const int lane = tid & 31;
        const int half = lane >> 4;   // 0: lanes 0-15, 1: lanes 16-31
        const int l15  = lane & 15;
        for (int tile = w; tile < NTILES; tile += NWAVES) {  // wave-uniform loop
            const int   col   = tile * 16 + l15;             // chunk index (B/D lane)
            const float cok   = (col < NCH) ? 1.0f : 0.0f;
            const int   cbase = ((col < NCH) ? col : 0) * 16;
            v8f acc = {0.f, 0.f, 0.f, 0.f, 0.f, 0.f, 0.f, 0.f};
#pragma unroll
            for (int kb = 0; kb < 4; ++kb) {
                // K index within wave32 layout: K = kb*4 + 2*half + vgpr
                const int k0 = kb * 4 + half * 2;
                v2f a, b;
                a.x = mcoef(l15, k0);             // A: lane = row M, vgpr/half = K
                a.y = mcoef(l15, k0 + 1);
                b.x = cok * xbuf[cbase + k0];     // B: lane = col N, vgpr/half = K
                b.y = cok * xbuf[cbase + k0 + 1];
                acc = __builtin_amdgcn_wmma_f32_16x16x4_f32(
                    /*neg_a=*/false, a, /*neg_b=*/false, b,
                    /*c_mod=*/(short)0, acc, /*reuse_a=*/false, /*reuse_b=*/false);
            }
            if (col < NCH) {
#pragma unroll
                for (int r = 0; r < 8; ++r)       // D: row = r + 8*half, col = lane
                    pbuf[col * 16 + (r + 8 * half)] = acc[r];
            }
        }
    }
    __syncthreads();

    // ---- Phase 3: scan of chunk carries  s[c] = A16*s[c-1] + partial[c][15] ----
    {
        const int G = (NCH + BLOCK - 1) / BLOCK;  // 3 chunks per thread
        float Aloc = 1.f, Bloc = 0.f;
        for (int g = 0; g < G; ++g) {
            const int c = tid * G + g;
            if (c < NCH) { Bloc = A16 * Bloc + pbuf[c * 16 + 15]; Aloc *= A16; }
        }
        scanA[tid] = Aloc; scanB[tid] = Bloc;
        __syncthreads();
        for (int off = 1; off < BLOCK; off <<= 1) {   // Hillis-Steele, 9 steps
            const float a0 = scanA[tid], b0 = scanB[tid];
            float pA = 1.f, pB = 0.f;
            if (tid >= off) { pA = scanA[tid - off]; pB = scanB[tid - off]; }
            __syncthreads();
            scanA[tid] = a0 * pA;
            scanB[tid] = a0 * pB + b0;
            __syncthreads();
        }
        float sc = (tid > 0) ? scanB[tid - 1] : 0.f;  // carry entering my range
        for (int g = 0; g < G; ++g) {
            const int c = tid * G + g;
            if (c < NCH) { sc = A16 * sc + pbuf[c * 16 + 15]; sarr[c] = sc; }
        }
    }
    __syncthreads();

    // ---- Phase 4: fixup + transient generator (overwrite xbuf) ----
    const float twopi = 6.283185307179586f;
    for (int i = tid; i < NS; i += BLOCK) {
        const float t     = (float)i * tstep;
        const int   row   = i & 15, c = i >> 4;
        const float carry = (c > 0) ? sarr[c - 1] : 0.f;
        const float filt  = pbuf[i] + powf(ADEC, (float)(row + 1)) * carry;
        float sine = __sinf(twopi * freq_t * t) * __expf(-decay_t * t) * gain_t;
        sine = tanhf(sine * sat);
        const float ncomp = filt * __expf(-(decay_t * 20.f) * t) * (gain_t * 0.5f);
        xbuf[i] = sine + ncomp;              // transient
    }
    __syncthreads();

    // ---- Phase 5: resonator wavefront (out lives in pbuf) ----
    const float delay = SRATE / freq_r;
    int C = (int)floorf(delay - 1.0f);       // chunk size: deps strictly earlier
    if (C < 1)  C = 1;
    if (C > NS) C = NS;
    if (tid == 0) pbuf[0] = xbuf[0];
    __syncthreads();
    for (int s = 1; s < NS; s += C) {
        const int e = (s + C < NS) ? (s + C) : NS;
        for (int i = s + tid; i < e; i += BLOCK) {
            const float dly = (float)i - delay;
            const float vm  = (dly >= 0.f) ? 1.f : 0.f;
            int fl = (int)floorf(dly);
            fl = fl < 0 ? 0 : (fl > NS - 1 ? NS - 1 : fl);
            const int   ce = (fl + 1 > NS - 1) ? NS - 1 : fl + 1;
            const float fr = dly - (float)fl;          // matches reference (post-clamp)
            const float interp = (1.f - fr) * pbuf[fl] + fr * pbuf[ce];
            pbuf[i] = xbuf[i] + feedback * interp * vm;
        }
        __syncthreads();
    }

    // ---- Phase 6: noise generator + final sum ----
    for (int i = tid; i < NS; i += BLOCK) {
        const float t   = (float)i * tstep;
        const float env = (1.f - __expf(-attack_n * t)) * __expf(-decay_n * t);
        out[i] = xbuf[i] + noise_n[i] * env * gain_n + pbuf[i] * gain_r;
    }
}

extern "C" void kernel_launch(void* const* d_in, const int* in_sizes, int n_in,
                              void* d_out, int out_size, void* d_ws, size_t ws_size,
                              hipStream_t stream) {
    (void)in_sizes; (void)n_in; (void)d_ws; (void)ws_size; (void)out_size;
    const float* params  = (const float*)d_in[0];
    const float* noise_t = (const float*)d_in[1];
    const float* noise_n = (const float*)d_in[2];
    float* out = (float*)d_out;

    const size_t shmem = (size_t)(2 * NS + NCH + 4 + 2 * BLOCK) * sizeof(float); // ~202 KB
    // CDNA5 WGP has 320KB LDS; raise the dynamic-LDS cap (deterministic,
    // non-stream attribute call — not graph work).
    hipFuncSetAttribute((const void*)drum_synth_kernel,
                        hipFuncAttributeMaxDynamicSharedMemorySize, (int)shmem);
    drum_synth_kernel<<<1, BLOCK, shmem, stream>>>(params, noise_t, noise_n, out);
}